// L_segexp_62294205661754
// MI455X (gfx1250) — compile-verified
//
#include <hip/hip_runtime.h>

// ---------------------------------------------------------------------------
// Segmented mean/variance stats over 35 bins, fused into one streaming pass.
//   traffic = 134 MB read-once  ->  HBM-bound, ~5.7 us floor @ 23.3 TB/s
//   stats binned per-block in LDS (conflict-free per-lane columns, ds_add_f32)
//   per-block lane-reduction done as M x ones via V_WMMA_F32_16X16X4_F32
//   per-block combine: 105 divergent global_atomic_add_f32 (no readlane scans)
// ---------------------------------------------------------------------------

#define NUM_BINS 35
#define NROWS    105            // 3 stats (sum, sumsq, count) * 35 bins
#define PROWS    112            // padded to 7 groups of 16 rows for WMMA
#define LCOLS    33             // 32 lane-columns + 1 pad (bank spread)
#define HW       (512 * 512)
#define NPIX     (32 * HW)      // 8,388,608 pixels
#define NG       (NPIX / 4)     // float4 groups

typedef float v2f __attribute__((ext_vector_type(2)));
typedef float v4f __attribute__((ext_vector_type(4)));
typedef float v8f __attribute__((ext_vector_type(8)));
typedef int   v4i __attribute__((ext_vector_type(4)));

__global__ void L_segexp_zero_ws(float* __restrict__ ws) {
    int i = threadIdx.x;
    if (i < NROWS) ws[i] = 0.0f;
}

__global__ __launch_bounds__(256)
void L_segexp_stats(const float* __restrict__ x,
                    const int*   __restrict__ y,
                    float*       __restrict__ ws) {
    __shared__ float smat[PROWS * LCOLS];           // 14.8 KB of 320 KB/WGP
    __shared__ float rres[PROWS];                   // per-block row sums

    const int tid  = threadIdx.x;
    const int lane = tid & 31;

    for (int i = tid; i < PROWS * LCOLS; i += blockDim.x) smat[i] = 0.0f;
    __syncthreads();

    // ---- streaming phase: 16 B/pixel, non-temporal (read-once data) ------
    const int stride = gridDim.x * blockDim.x;
    for (int g = blockIdx.x * blockDim.x + tid; g < NG; g += stride) {
        const int b  = g >> 16;                 // g / (HW/4), HW/4 = 65536
        const int i4 = g & 65535;
        const float* xb = x + ((long long)b * (3 * HW) + (long long)i4 * 4);

        v4f x0 = __builtin_nontemporal_load((const v4f*)(xb));
        v4f x1 = __builtin_nontemporal_load((const v4f*)(xb + HW));
        v4f x2 = __builtin_nontemporal_load((const v4f*)(xb + 2 * HW));
        v4i yv = __builtin_nontemporal_load((const v4i*)y + g);

        v4f xm = (x0 + x1 + x2) * (1.0f / 3.0f);   // channel mean

        #pragma unroll
        for (int k = 0; k < 4; ++k) {
            const float v   = xm[k];
            const int   bin = yv[k];
            // column == lane: no intra-wave address collisions -> ds_add_f32
            // proceeds with zero atomic serialization.
            atomicAdd(&smat[(      bin) * LCOLS + lane], v);
            atomicAdd(&smat[(35 + bin) * LCOLS + lane], v * v);
            atomicAdd(&smat[(70 + bin) * LCOLS + lane], 1.0f);
        }
    }
    __syncthreads();

    // ---- reduction phase: row-sums of smat[PROWS][32] == smat x ones -----
    // wave 0 only; EXEC is all-ones for the WMMAs (wave-uniform branch).
    if (tid < 32) {
        const int mrow = lane & 15;             // A-matrix row  (M = lane%16)
        const int colb = (lane >> 4) << 1;      // A-matrix K lsb (0 or 2)
        const v2f bones = {1.0f, 1.0f};         // B = ones(4x16), all lanes

        for (int grp = 0; grp < PROWS / 16; ++grp) {
            const int r0 = grp * 16;
            v8f c = {};
            #pragma unroll
            for (int kb = 0; kb < 32; kb += 4) { // chain 8 WMMAs over K
                const float* p = &smat[(r0 + mrow) * LCOLS + kb + colb];
                v2f a;
                a.x = p[0];                     // VGPR0: K = kb + colb
                a.y = p[1];                     // VGPR1: K = kb + colb + 1
                c = __builtin_amdgcn_wmma_f32_16x16x4_f32(
                        false, a, false, bones, (short)0, c, false, false);
            }
            // Every column of D equals the row sum (B = ones).  Lane 0 holds
            // rows r0..r0+7 in c[0..7]; lane 16 holds rows r0+8..r0+15.
            // Spill to LDS (divergent ds_store; no atomic-optimizer scans).
            if (lane == 0)  *(v8f*)&rres[r0]     = c;
            if (lane == 16) *(v8f*)&rres[r0 + 8] = c;
        }
    }
    __syncthreads();

    // ---- combine: one divergent-address f32 atomic per row ---------------
    if (tid < NROWS) atomicAdd(&ws[tid], rres[tid]);
}

__global__ void L_segexp_finalize(const float* __restrict__ ws,
                                  float*       __restrict__ out) {
    if (threadIdx.x == 0 && blockIdx.x == 0) {
        const float n = (float)NPIX;            // reference divides by FULL n
        float total = 0.0f;
        for (int b = 0; b < NUM_BINS; ++b) {
            const float s  = ws[b];
            const float q  = ws[35 + b];
            const float cn = ws[70 + b];
            const float d2 = s / n;
            total += (q - 2.0f * d2 * s + cn * d2 * d2) / n;
        }
        out[0] = total;
    }
}

extern "C" void kernel_launch(void* const* d_in, const int* in_sizes, int n_in,
                              void* d_out, int out_size, void* d_ws, size_t ws_size,
                              hipStream_t stream) {
    const float* x  = (const float*)d_in[0];   // [32,3,512,512] f32
    const int*   y  = (const int*)  d_in[1];   // [32,512,512]   i32
    float*       ws = (float*)d_ws;            // 105 f32 accumulators
    float*       out = (float*)d_out;          // scalar f32

    L_segexp_zero_ws<<<1, 128, 0, stream>>>(ws);
    // 2048 blocks x 256 thr = 524,288 lanes; NG / lanes = 4 float4-groups each.
    L_segexp_stats<<<2048, 256, 0, stream>>>(x, y, ws);
    L_segexp_finalize<<<1, 32, 0, stream>>>(ws, out);
}